// ds_us_43009802502566
// MI455X (gfx1250) — compile-verified
//
#include <hip/hip_runtime.h>
#include <stdint.h>

// Problem constants (from the reference).
#define B_DIM  32
#define C_DIM  64
#define N_IN   10475
#define N_OUT  2619
#define NNZ_T  (8 * N_OUT)          // 20952
#define BC     (B_DIM * C_DIM)      // 2048 channels
#define G_CH   4                    // channels per workgroup (LDS-resident x rows)
#define NBLK   (BC / G_CH)          // 512 workgroups
#define NTHR   256                  // 8 wave32s
#define CHUNK  512                  // COO entries staged per chunk
#define NCHUNK ((NNZ_T + CHUNK - 1) / CHUNK)   // 41 (tail = 472, even)
#define XF4    ((G_CH * N_IN) / 4)  // 10475 float4 groups (exact: 41900 % 4 == 0)

// ---- CDNA5 async global<->LDS helpers (gfx1250) -------------------------

// Generic pointer (into __shared__) -> 32-bit LDS byte offset.
__device__ __forceinline__ unsigned lds_off(const void* p) {
  return (unsigned)(unsigned long long)(__attribute__((address_space(3))) const void*)p;
}

__device__ __forceinline__ void async_load_b64(unsigned lds, const void* gaddr) {
  asm volatile("global_load_async_to_lds_b64 %0, %1, off"
               :: "v"(lds), "v"(gaddr) : "memory");
}
__device__ __forceinline__ void async_load_b128(unsigned lds, const void* gaddr) {
  asm volatile("global_load_async_to_lds_b128 %0, %1, off"
               :: "v"(lds), "v"(gaddr) : "memory");
}
__device__ __forceinline__ void async_store_b128(void* gaddr, unsigned lds) {
  asm volatile("global_store_async_from_lds_b128 %0, %1, off"
               :: "v"(gaddr), "v"(lds) : "memory");
}
__device__ __forceinline__ void wait_async_le3() {
  asm volatile("s_wait_asynccnt 0x3" ::: "memory");
}
__device__ __forceinline__ void wait_async_0() {
  asm volatile("s_wait_asynccnt 0x0" ::: "memory");
}

// ---- Kernel -------------------------------------------------------------

__global__ __launch_bounds__(NTHR) void spmm_bary_kernel(
    const float* __restrict__ x,      // [BC, N_IN]
    const int*   __restrict__ rows,   // [NNZ]
    const int*   __restrict__ cols,   // [NNZ]
    const float* __restrict__ vals,   // [NNZ]
    float*       __restrict__ out)    // [BC, N_OUT]
{
  // CDNA5: 320 KB LDS per WGP. Total here: 221,792 B (one WG per WGP).
  __shared__ __align__(16) float s_x[G_CH * N_IN];         // 167600 B: x rows
  __shared__ __align__(16) float acc[G_CH * N_OUT];        //  41904 B: output acc
  __shared__ __align__(16) int   s_rows[2][CHUNK];         //   4096 B
  __shared__ __align__(16) int   s_cols[2][CHUNK];         //   4096 B
  __shared__ __align__(16) float s_vals[2][CHUNK];         //   4096 B

  const int tid   = threadIdx.x;
  const int chan0 = blockIdx.x * G_CH;
  const float* xb = x + (size_t)chan0 * N_IN;   // 16B-aligned: chan0*N_IN*4 = 167600*blk

  // (1) Stream this WG's 4 x-rows straight into LDS with async b128 DMAs:
  //     coalesced, single pass over x, no VGPR round trip, ~41 in flight/wave.
  for (int i4 = tid; i4 < XF4; i4 += NTHR)
    async_load_b128(lds_off(&s_x[4 * i4]), (const char*)xb + 16 * (size_t)i4);

  // (2) Zero accumulators (plain ds stores; barrier before first use below).
  for (int i = tid; i < G_CH * N_OUT; i += NTHR) acc[i] = 0.0f;

  // Stage one COO chunk (rows/cols/vals) into buffer `bsel` with async b64
  // DMAs. Every wave issues exactly 3 loads per call (tail clamped to the
  // last even pair; duplicate lanes rewrite identical data -> benign), so
  // per-wave ASYNCcnt accounting is uniform across the block.
  auto stage = [&](int chunk, int bsel) {
    const int base = chunk * CHUNK;
    const int cnt  = (NNZ_T - base < CHUNK) ? (NNZ_T - base) : CHUNK;
    int e = 2 * tid;
    if (e > cnt - 2) e = cnt - 2;          // cnt always even
    async_load_b64(lds_off(&s_rows[bsel][e]), rows + base + e);
    async_load_b64(lds_off(&s_cols[bsel][e]), cols + base + e);
    async_load_b64(lds_off(&s_vals[bsel][e]), vals + base + e);
  };

  stage(0, 0);   // outstanding per wave: 41 (x-fill) + 3 (chunk 0)

  for (int ch = 0; ch < NCHUNK; ++ch) {
    // Prefetch next chunk into the other buffer (restage the final chunk on
    // the last iteration so every iteration issues exactly 3 loads/wave).
    const int nb = (ch + 1 < NCHUNK) ? (ch + 1) : ch;
    stage(nb, (ch + 1) & 1);
    wait_async_le3();      // in-order loads => x-fill (iter 0) + chunk ch done
    __syncthreads();       // staged data + s_x visible to all waves

    const int base = ch * CHUNK;
    const int cnt  = (NNZ_T - base < CHUNK) ? (NNZ_T - base) : CHUNK;
    const int bsel = ch & 1;

    for (int k = tid; k < cnt; k += NTHR) {
      const int   r = s_rows[bsel][k];
      const int   c = s_cols[bsel][k];
      const float v = s_vals[bsel][k];
#pragma unroll
      for (int g = 0; g < G_CH; ++g) {
        const float xv = s_x[g * N_IN + c];            // ds_load: LDS gather
        atomicAdd(&acc[g * N_OUT + r], v * xv);        // ds_add_f32: LDS scatter
      }
    }
    __syncthreads();   // buffer (ch&1) is overwritten next iteration
  }

  // (3) Drain leftover DMAs, then stream the contiguous 41,904-byte
  //     accumulator block (16*2619) LDS -> global with async b128 stores.
  wait_async_0();
  __syncthreads();

  char*       gdst  = (char*)(out + (size_t)chan0 * N_OUT);  // blk*41904: 16B aligned
  const char* lsrc  = (const char*)acc;
  const int   total = G_CH * N_OUT * (int)sizeof(float);     // 41904
  for (int ofs = tid * 16; ofs < total; ofs += NTHR * 16)
    async_store_b128(gdst + ofs, lds_off(lsrc + ofs));

  wait_async_0();   // stores complete (s_endpgm also waits idle)
}

// ---- Host-side launch ---------------------------------------------------

extern "C" void kernel_launch(void* const* d_in, const int* in_sizes, int n_in,
                              void* d_out, int out_size, void* d_ws, size_t ws_size,
                              hipStream_t stream) {
  (void)in_sizes; (void)n_in; (void)out_size; (void)d_ws; (void)ws_size;
  const float* x    = (const float*)d_in[0];
  const int*   rows = (const int*)d_in[1];
  const int*   cols = (const int*)d_in[2];
  const float* vals = (const float*)d_in[3];
  float*       out  = (float*)d_out;
  spmm_bary_kernel<<<NBLK, NTHR, 0, stream>>>(x, rows, cols, vals, out);
}